// RegressionLoss_231928234571
// MI455X (gfx1250) — compile-verified
//
#include <hip/hip_runtime.h>
#include <hip/hip_bf16.h>
#include <math.h>

typedef __attribute__((ext_vector_type(2))) float v2f;
typedef __attribute__((ext_vector_type(8))) float v8f;

#define NBAT 16
#define NQ   2048
#define NK   1024
#define NG   512
#define NM   4096
#define BLK  256
#define NBLK (NM / BLK)   // 16 blocks -> 16x4 partial matrix, one f32 WMMA tile

__device__ __forceinline__ float wave_sum(float v) {
    #pragma unroll
    for (int off = 16; off > 0; off >>= 1) v += __shfl_down(v, off, 32);
    return v;
}

// point-in-rotated-box test (reference: _points_in_box)
__device__ __forceinline__ bool pt_in_box(float x, float y, const float* bx) {
    float cs = cosf(bx[6]), sn = sinf(bx[6]);
    float dx = x - bx[0], dy = y - bx[1];
    float qx = dx * cs + dy * sn;
    float qy = -dx * sn + dy * cs;
    return (fabsf(qx) <= bx[3] * 0.5f + 1e-5f) && (fabsf(qy) <= bx[4] * 0.5f + 1e-5f);
}

__device__ __forceinline__ void bev_corners(const float* bx, float c[4][2]) {
    float cs = cosf(bx[6]), sn = sinf(bx[6]);
    float hx = 0.5f * bx[3], hy = 0.5f * bx[4];
    const float lx[4] = { hx, hx, -hx, -hx };
    const float ly[4] = { hy, -hy, -hy, hy };
    #pragma unroll
    for (int i = 0; i < 4; ++i) {
        c[i][0] = bx[0] + lx[i] * cs - ly[i] * sn;
        c[i][1] = bx[1] + lx[i] * sn + ly[i] * cs;
    }
}

__global__ __launch_bounds__(BLK) void pair_loss_kernel(
    const float* __restrict__ pred_boxes, const float* __restrict__ pred_ious,
    const float* __restrict__ gt_boxes,   const int* __restrict__ topk,
    const int* __restrict__ batch_idx,    const int* __restrict__ src_idx,
    const int* __restrict__ tgt_idx,      float* __restrict__ ws_part)
{
    const int m = blockIdx.x * BLK + threadIdx.x;
    const int b = batch_idx[m];
    const int s = src_idx[m];
    const int t = tgt_idx[m];
    const int k = topk[b * NK + s];

    const float* pbp = pred_boxes + ((size_t)b * NQ + k) * 7;
    const float* gbp = gt_boxes   + ((size_t)b * NG + t) * 7;
    const float  pim = pred_ious[(size_t)b * NQ + k];

    float p[7], g[7];
    #pragma unroll
    for (int j = 0; j < 7; ++j) { p[j] = pbp[j]; g[j] = gbp[j]; }

    // ---- L1 losses (raw units) ----
    float l_bbox = 0.f;
    #pragma unroll
    for (int j = 0; j < 6; ++j) l_bbox += fabsf(p[j] - g[j]);
    float l_rad = fabsf(p[6] - g[6]);

    // ---- axis-aligned 3D GIoU on raw boxes (diagonal of reference matrix) ----
    float lo1[3], hi1[3], lo2[3], hi2[3];
    #pragma unroll
    for (int j = 0; j < 3; ++j) {
        lo1[j] = p[j] - 0.5f * p[j + 3]; hi1[j] = p[j] + 0.5f * p[j + 3];
        lo2[j] = g[j] - 0.5f * g[j + 3]; hi2[j] = g[j] + 0.5f * g[j + 3];
    }
    float vol1 = (hi1[0]-lo1[0]) * (hi1[1]-lo1[1]) * (hi1[2]-lo1[2]);
    float vol2 = (hi2[0]-lo2[0]) * (hi2[1]-lo2[1]) * (hi2[2]-lo2[2]);
    float inter = 1.f, volc = 1.f;
    #pragma unroll
    for (int j = 0; j < 3; ++j) {
        inter *= fmaxf(fminf(hi1[j], hi2[j]) - fmaxf(lo1[j], lo2[j]), 0.f);
        volc  *= fmaxf(fmaxf(hi1[j], hi2[j]) - fminf(lo1[j], lo2[j]), 0.f);
    }
    float uni  = vol1 + vol2 - inter;
    float iou  = inter / fmaxf(uni, 1e-7f);
    float giou = iou - (volc - uni) / fmaxf(volc, 1e-7f);
    float l_giou = 1.f - giou;

    // ---- decode boxes to physical units ----
    float A[7], Bb[7];
    A[0]  = p[0] * 150.4f - 75.2f;  Bb[0] = g[0] * 150.4f - 75.2f;
    A[1]  = p[1] * 150.4f - 75.2f;  Bb[1] = g[1] * 150.4f - 75.2f;
    A[2]  = p[2] * 6.0f   - 2.0f;   Bb[2] = g[2] * 6.0f   - 2.0f;
    A[3]  = p[3] * 20.f   + 0.05f;  Bb[3] = g[3] * 20.f   + 0.05f;
    A[4]  = p[4] * 20.f   + 0.05f;  Bb[4] = g[4] * 20.f   + 0.05f;
    A[5]  = p[5] * 8.f    + 0.05f;  Bb[5] = g[5] * 8.f    + 0.05f;
    A[6]  = p[6] * 6.28318530717958647692f - 3.14159265358979323846f;
    Bb[6] = g[6] * 6.28318530717958647692f - 3.14159265358979323846f;

    // ---- rotated BEV polygon intersection IoU3D ----
    float ca[4][2], cb[4][2];
    bev_corners(A, ca);
    bev_corners(Bb, cb);

    float px[24], py[24], ang[24];
    bool  mk[24];
    #pragma unroll
    for (int i = 0; i < 4; ++i) {
        px[i] = ca[i][0]; py[i] = ca[i][1]; mk[i] = pt_in_box(px[i], py[i], Bb);
        px[4+i] = cb[i][0]; py[4+i] = cb[i][1]; mk[4+i] = pt_in_box(px[4+i], py[4+i], A);
    }
    #pragma unroll
    for (int i = 0; i < 4; ++i) {
        float a0x = ca[i][0], a0y = ca[i][1];
        float dax = ca[(i+1)&3][0] - a0x, day = ca[(i+1)&3][1] - a0y;
        #pragma unroll
        for (int j = 0; j < 4; ++j) {
            float e0x = cb[j][0], e0y = cb[j][1];
            float dbx = cb[(j+1)&3][0] - e0x, dby = cb[(j+1)&3][1] - e0y;
            float dpx = e0x - a0x, dpy = e0y - a0y;
            float den = dax * dby - day * dbx;
            bool dok  = fabsf(den) > 1e-8f;
            float dens = dok ? den : 1.f;
            float tt = (dpx * dby - dpy * dbx) / dens;
            float uu = (dpx * day - dpy * dax) / dens;
            bool ok = dok && (tt >= 0.f) && (tt <= 1.f) && (uu >= 0.f) && (uu <= 1.f);
            int idx = 8 + i * 4 + j;
            px[idx] = a0x + tt * dax;
            py[idx] = a0y + tt * day;
            mk[idx] = ok;
        }
    }
    float cnt = 0.f, sx = 0.f, sy = 0.f;
    #pragma unroll
    for (int i = 0; i < 24; ++i) {
        float w = mk[i] ? 1.f : 0.f;
        cnt += w; sx += px[i] * w; sy += py[i] * w;
    }
    float cd = fmaxf(cnt, 1.f);
    float cx = sx / cd, cy = sy / cd;
    #pragma unroll
    for (int i = 0; i < 24; ++i)
        ang[i] = mk[i] ? atan2f(py[i] - cy, px[i] - cx) : 1e9f;

    // stable insertion sort on angle (matches stable jnp.argsort)
    for (int i = 1; i < 24; ++i) {
        float ka = ang[i], kx = px[i], ky = py[i];
        bool km = mk[i];
        int j = i - 1;
        while (j >= 0 && ang[j] > ka) {
            ang[j+1] = ang[j]; px[j+1] = px[j]; py[j+1] = py[j]; mk[j+1] = mk[j];
            --j;
        }
        ang[j+1] = ka; px[j+1] = kx; py[j+1] = ky; mk[j+1] = km;
    }
    float p0x = px[0], p0y = py[0];
    for (int i = 0; i < 24; ++i)
        if (!mk[i]) { px[i] = p0x; py[i] = p0y; }
    float acc = 0.f;
    for (int i = 0; i < 24; ++i) {
        int j = (i + 1 < 24) ? i + 1 : 0;
        acc += px[i] * py[j] - px[j] * py[i];
    }
    float area = 0.5f * fabsf(acc);
    float inter_bev = (cnt >= 3.f) ? area : 0.f;
    float zlo = fmaxf(A[2] - 0.5f * A[5], Bb[2] - 0.5f * Bb[5]);
    float zhi = fminf(A[2] + 0.5f * A[5], Bb[2] + 0.5f * Bb[5]);
    float vinter = inter_bev * fmaxf(zhi - zlo, 0.f);
    float v1 = A[3] * A[4] * A[5];
    float v2 = Bb[3] * Bb[4] * Bb[5];
    float iou3 = vinter / fmaxf(v1 + v2 - vinter, 1e-7f);
    float l_iou = fabsf(pim - (2.f * iou3 - 1.f));

    // ---- block reduction: wave shfl -> LDS float atomics (ds_add_f32) ----
    __shared__ float ssum[4];
    if (threadIdx.x < 4) ssum[threadIdx.x] = 0.f;
    __syncthreads();
    float r0 = wave_sum(l_bbox);
    float r1 = wave_sum(l_giou);
    float r2 = wave_sum(l_rad);
    float r3 = wave_sum(l_iou);
    if ((threadIdx.x & 31) == 0) {
        atomicAdd(&ssum[0], r0);
        atomicAdd(&ssum[1], r1);
        atomicAdd(&ssum[2], r2);
        atomicAdd(&ssum[3], r3);
    }
    __syncthreads();
    if (threadIdx.x < 4)
        ws_part[blockIdx.x * 4 + threadIdx.x] = ssum[threadIdx.x];
}

// Final reduction: the 16(blocks) x 4(components) fp32 partial matrix is summed
// along the block axis with one V_WMMA_F32_16X16X4_F32, A = all-ones:
//   D[m,n] = sum_k B[k,n]  (exact fp32 column sums, independent of B row order).
// Column n encodes (component c = n&3, block-group g = n>>2); three shfl adds
// combine the 4 groups. One wave, EXEC all ones before the WMMA.
__global__ __launch_bounds__(32) void reduce_wmma_kernel(
    const float* __restrict__ part, const int* __restrict__ nbp,
    float* __restrict__ out)
{
    const int l = threadIdx.x;           // 0..31, full wave32
    const int col = l & 15;
    const int c = col & 3;
    const int g = col >> 2;
    const int rbase = (l < 16) ? 0 : 2;  // two B rows per lane-half

    v2f a; a[0] = 1.f; a[1] = 1.f;       // A = ones(16x4)
    v2f bv;
    bv[0] = part[(4 * g + rbase + 0) * 4 + c];
    bv[1] = part[(4 * g + rbase + 1) * 4 + c];
    v8f cz = {};
    v8f d = __builtin_amdgcn_wmma_f32_16x16x4_f32(
        /*neg_a=*/false, a, /*neg_b=*/false, bv,
        /*c_mod=*/(short)0, cz, /*reuse_a=*/false, /*reuse_b=*/false);

    float v = d[0];                      // colsum(col) in every lane
    int cc = l & 3;
    float tot = __shfl(v, cc, 32) + __shfl(v, cc + 4, 32) +
                __shfl(v, cc + 8, 32) + __shfl(v, cc + 12, 32);
    if (l < 4) out[l] = tot / (float)(*nbp);
}

extern "C" void kernel_launch(void* const* d_in, const int* in_sizes, int n_in,
                              void* d_out, int out_size, void* d_ws, size_t ws_size,
                              hipStream_t stream) {
    (void)in_sizes; (void)n_in; (void)out_size; (void)ws_size;
    const float* pred_boxes = (const float*)d_in[0];
    const float* pred_ious  = (const float*)d_in[1];
    const float* gt_boxes   = (const float*)d_in[2];
    const int*   topk       = (const int*)d_in[3];
    const int*   batch_idx  = (const int*)d_in[4];
    const int*   src_idx    = (const int*)d_in[5];
    const int*   tgt_idx    = (const int*)d_in[6];
    const int*   num_boxes  = (const int*)d_in[7];
    float* part = (float*)d_ws;          // 16 blocks x 4 floats

    pair_loss_kernel<<<NBLK, BLK, 0, stream>>>(
        pred_boxes, pred_ious, gt_boxes, topk, batch_idx, src_idx, tgt_idx, part);
    reduce_wmma_kernel<<<1, 32, 0, stream>>>(part, num_boxes, (float*)d_out);
}